// RubiksShift2D_91010357002308
// MI455X (gfx1250) — compile-verified
//
#include <hip/hip_runtime.h>
#include <stdint.h>

// Problem shape (fixed by the reference)
#define NN 32
#define CC 256
#define HH 56
#define WW 56
#define PLANE (HH * WW)          // 3136 floats = 12544 bytes, contiguous per (n,c)
#define PLANES_PER_BLOCK 8
#define THREADS 256              // 8 wave32s

typedef __attribute__((ext_vector_type(4))) unsigned int v4u;
typedef __attribute__((ext_vector_type(8))) int          v8i;
typedef __attribute__((ext_vector_type(4))) int          v4i;

static __device__ __forceinline__ unsigned sread(unsigned v) {
  return (unsigned)__builtin_amdgcn_readfirstlane((int)v);
}

// Issue one TDM descriptor: DMA a contiguous 3136-float plane from global
// memory into LDS at byte offset lds_byte_off. 1-D tile, data_size = 4 bytes.
// D# layout per CDNA5 ISA ch. 7/8 (async_tensor doc §8.3/§8.4).
static __device__ __forceinline__ void tdm_load_plane(const float* gsrc,
                                                      unsigned lds_byte_off) {
  const uint64_t ga = (uint64_t)(uintptr_t)gsrc;

  v4u g0;
  g0.x = 1u;                                        // count=1 (valid user D#)
  g0.y = sread(lds_byte_off);                       // lds_addr (bytes)
  g0.z = sread((unsigned)(ga & 0xFFFFFFFFu));       // global_addr[31:0]
  g0.w = sread(((unsigned)(ga >> 32) & 0x01FFFFFFu) // global_addr[56:32]
               | (2u << 30));                       // type=2 ("image")

  v8i g1;
  g1[0] = (int)(2u << 16);                 // wg_mask=0, data_size=2 (4B), no pad/iter
  g1[1] = (int)((PLANE & 0xFFFFu) << 16);  // atomic_barrier=0; tensor_dim0 lo16
  g1[2] = (int)(1u << 16);                 // tensor_dim0 hi16 = 0; tensor_dim1 = 1
  g1[3] = (int)((unsigned)PLANE << 16);    // tensor_dim1 hi = 0; tile_dim0 = 3136
  g1[4] = 0;                               // tile_dim1 = 0, tile_dim2 = 0 (1-D tile)
  g1[5] = PLANE;                           // tensor_dim0_stride lo32
  g1[6] = 0;                               // stride hi, tensor_dim1_stride lo
  g1[7] = 0;

  v4i z4 = {0, 0, 0, 0};                   // groups 2/3: dims/strides unused (0)
#if defined(__clang_major__) && (__clang_major__ >= 23)
  v8i z8 = {0, 0, 0, 0, 0, 0, 0, 0};
  __builtin_amdgcn_tensor_load_to_lds(g0, g1, z4, z4, z8, 0);
#else
  __builtin_amdgcn_tensor_load_to_lds(g0, g1, z4, z4, 0);
#endif
}

__global__ __launch_bounds__(THREADS)
void RubiksShift2D_kernel(const float* __restrict__ x,
                          const float* __restrict__ shift,
                          float* __restrict__ out) {
  __shared__ float smem[2][PLANE];        // double buffer: 2 x 12544 B = 25 KB

  const int tid    = threadIdx.x;
  const int plane0 = blockIdx.x * PLANES_PER_BLOCK;
  const unsigned lds_base = (unsigned)(uintptr_t)&smem[0][0]; // low 32b = LDS offset
  const bool issuer = (tid < 32);         // wave 0 drives the TDM pipeline

  // Prime the pipeline: plane 0 -> buffer 0.
  if (issuer) {
    tdm_load_plane(x + (size_t)plane0 * PLANE, lds_base);
  }

  for (int k = 0; k < PLANES_PER_BLOCK; ++k) {
    const int p = plane0 + k;
    const int c = p & (CC - 1);

    if (issuer) {
      if (k + 1 < PLANES_PER_BLOCK) {
        // Prefetch next plane into the other buffer, then wait until the
        // *current* plane's DMA is done (<=1 outstanding tensor op).
        tdm_load_plane(x + (size_t)(p + 1) * PLANE,
                       lds_base + (unsigned)(((k + 1) & 1) * PLANE * 4));
        __builtin_amdgcn_s_wait_tensorcnt(1);
      } else {
        __builtin_amdgcn_s_wait_tensorcnt(0);
      }
    }
    __syncthreads();                      // plane k now visible in LDS to all waves

    const float* buf = smem[k & 1];

    // Per-channel bilinear weights (uniform across the block).
    const float a  = shift[c];
    const float b  = shift[CC + c];
    const float fa = floorf(a);
    const float fb = floorf(b);
    const int   ia = (int)fa;
    const int   ib = (int)fb;
    const float wa = a - fa;
    const float wb = b - fb;
    const float w00 = (1.0f - wa) * (1.0f - wb);
    const float w01 = (1.0f - wa) * wb;
    const float w10 = wa * (1.0f - wb);
    const float w11 = wa * wb;

    float* op = out + (size_t)p * PLANE;

    // 56 % 4 == 0: every aligned float4 group stays within one row.
    const int GROUPS = PLANE / 4;         // 784
    for (int t = tid; t < GROUPS; t += THREADS) {
      const int i  = t / (WW / 4);        // row
      const int g  = t - i * (WW / 4);
      const int j0 = g * 4;

      const int r0 = i + ia;
      const int r1 = r0 + 1;
      const bool vr0 = ((unsigned)r0 < (unsigned)HH);
      const bool vr1 = ((unsigned)r1 < (unsigned)HH);
      const float* row0 = buf + (vr0 ? r0 : 0) * WW;
      const float* row1 = buf + (vr1 ? r1 : 0) * WW;

      float4 o;
      float* oe = (float*)&o;
#pragma unroll
      for (int e = 0; e < 4; ++e) {
        const int c0 = j0 + e + ib;
        const int c1 = c0 + 1;
        const bool vc0 = ((unsigned)c0 < (unsigned)WW);
        const bool vc1 = ((unsigned)c1 < (unsigned)WW);
        const int cc0 = vc0 ? c0 : 0;
        const int cc1 = vc1 ? c1 : 0;
        const float t00 = row0[cc0];
        const float t01 = row0[cc1];
        const float t10 = row1[cc0];
        const float t11 = row1[cc1];
        const float v00 = (vr0 && vc0) ? t00 : 0.0f;
        const float v01 = (vr0 && vc1) ? t01 : 0.0f;
        const float v10 = (vr1 && vc0) ? t10 : 0.0f;
        const float v11 = (vr1 && vc1) ? t11 : 0.0f;
        oe[e] = w00 * v00 + w01 * v01 + w10 * v10 + w11 * v11;
      }
      ((float4*)op)[t] = o;               // coalesced global_store_b128
    }
    __syncthreads();  // all reads of buf[k&1] done before it is reloaded at k+2
  }
}

extern "C" void kernel_launch(void* const* d_in, const int* in_sizes, int n_in,
                              void* d_out, int out_size, void* d_ws, size_t ws_size,
                              hipStream_t stream) {
  const float* x     = (const float*)d_in[0];   // [32, 256, 56, 56] fp32
  const float* shift = (const float*)d_in[1];   // [2, 256] fp32
  float*       out   = (float*)d_out;

  const int total_planes = NN * CC;             // 8192
  dim3 grid(total_planes / PLANES_PER_BLOCK);   // 1024 blocks
  dim3 block(THREADS);                          // 256 threads = 8 wave32
  RubiksShift2D_kernel<<<grid, block, 0, stream>>>(x, shift, out);
}